// BayesianDTW_86397562127159
// MI455X (gfx1250) — compile-verified
//
#include <hip/hip_runtime.h>
#include <stdint.h>

// Bayesian DTW on MI455X (gfx1250):
//   mu  : [64,385,385] wavefront log-sum-exp DP (sequential along anti-diagonals)
//   pi  : [64,384,384,3] 3-way softmax of mu neighbors (alpha*W cancels) * mask
//
// Kernel 1: one block per batch (64 blocks x 384 thr = 12 waves), anti-diagonal
//   wavefront. W is staged per-lane via CDNA5 async-to-LDS ring
//   (global_load_async_to_lds_b32 th:NT + s_wait_asynccnt). The workgroup
//   barrier is split (s_barrier_signal/-wait -1) so lane-private W staging
//   overlaps barrier skew instead of extending the 767-step critical path.
// Kernel 2: fully parallel softmax; pi streamed with non-temporal stores so
//   the 113 MB write stream does not evict mu from the 192 MB L2.

#define NEGV  (-1e20f)
#define NA    384
#define NB    384
#define BATCH 64
#define MU_W  (NB + 1)   /* 385 */
#define MU_H  (NA + 1)   /* 385 */
#define RING  32
#define LOOKA 24         /* async lookahead (<=25 outstanding < 63 ASYNCcnt max) */

__device__ __forceinline__ void async_load_w(unsigned lds_off, const float* gptr) {
    unsigned long long ga = (unsigned long long)(uintptr_t)gptr;
    // GLOBAL_LOAD_ASYNC_TO_LDS_B32: vdst = LDS byte address, vaddr = 64-bit global
    // W is consumed exactly once -> non-temporal.
    asm volatile("global_load_async_to_lds_b32 %0, %1, off th:TH_LOAD_NT"
                 :: "v"(lds_off), "v"(ga) : "memory");
}

// Split workgroup barrier (CDNA5 ISA: barrier ID -1 = workgroup).
// signal: drain DScnt first so our LDS writes are visible to waves that pass
// their wait after seeing our signal.
__device__ __forceinline__ void wg_barrier_signal() {
    asm volatile("s_wait_dscnt 0x0\n\t"
                 "s_barrier_signal -1" ::: "memory");
}
__device__ __forceinline__ void wg_barrier_wait() {
    asm volatile("s_barrier_wait -1" ::: "memory");
}

__global__ __launch_bounds__(NA) void dtw_mu_kernel(const float* __restrict__ W,
                                                    float* __restrict__ mu) {
    const int b = blockIdx.x;
    const int t = threadIdx.x;                       // owns mu row i = t+1
    const float* Wrow = W + (size_t)b * NA * NB + (size_t)t * NB;
    float* mub = mu + (size_t)b * MU_H * MU_W;

    __shared__ float ring[RING][NA];                 // 48 KB W staging ring
    __shared__ float dbuf[3][MU_W];                  // rotating diagonal buffers

    // ---- global boundary init: mu[0][:] and mu[:][0] ----
    mub[t] = (t == 0) ? 0.0f : NEGV;                 // row 0, cols 0..383
    if (t == 0) mub[NB] = NEGV;                      // row 0, col 384
    mub[(size_t)(t + 1) * MU_W] = NEGV;              // col 0, rows 1..384

    // ---- diagonal buffers: d1 = diag s=1, d2 = diag s=0 ----
    float* cur = dbuf[0];
    float* d1  = dbuf[1];
    float* d2  = dbuf[2];
    d1[t] = NEGV;                                    // mu[0][1], mu[1][0], pad
    d2[t] = NEGV;
    if (t == 0) { d1[NA] = NEGV; d2[NA] = NEGV; d2[0] = 0.0f; }  // mu[0][0] = 0

    // ---- prefill async ring for steps s = 2 .. 2+LOOKA-1 ----
    #pragma unroll 4
    for (int k = 0; k < LOOKA; ++k) {
        int s = 2 + k;
        int j = s - t - 2;
        j = (j < 0) ? 0 : ((j > NB - 1) ? NB - 1 : j);   // clamp: uniform ASYNCcnt
        async_load_w((unsigned)(uintptr_t)&ring[s % RING][t], Wrow + j);
    }

    wg_barrier_signal();                             // publishes LDS init

    // ---- 767 anti-diagonal wavefront steps ----
    for (int s = 2; s <= NA + NB; ++s) {
        // ==== lane-private region (overlaps barrier skew) ====
        {
            int sf = s + LOOKA;                      // prefetch LOOKA steps ahead
            int j = sf - t - 2;
            j = (j < 0) ? 0 : ((j > NB - 1) ? NB - 1 : j);
            async_load_w((unsigned)(uintptr_t)&ring[sf % RING][t], Wrow + j);
        }
        asm volatile("s_wait_asynccnt %0" :: "n"(LOOKA) : "memory");
        float w = ring[s % RING][t];                 // W[t][s-t-2] (own lane's data)

        // ==== cross-wave region: strictly between wait and signal ====
        wg_barrier_wait();                           // d1/d2 of step s-1 visible;
                                                     // safe to recycle cur buffer
        if (t == 0) {
            cur[0] = NEGV;                           // mu[0][s]
            if (s <= NA) cur[s] = NEGV;              // mu[s][0]
        }
        int jm = s - t - 1;                          // mu column for row i=t+1
        if (jm >= 1 && jm <= NB) {
            float up   = d1[t];                      // mu[i-1][jm]
            float left = d1[t + 1];                  // mu[i][jm-1]
            float dg   = d2[t];                      // mu[i-1][jm-1]
            float m  = fmaxf(up, fmaxf(left, dg));
            float sm = __expf(up - m) + __expf(left - m) + __expf(dg - m);
            float val = w + m + __logf(sm);
            cur[t + 1] = val;
            mub[(size_t)(t + 1) * MU_W + jm] = val;
        }
        wg_barrier_signal();                         // cur of step s published

        float* tmp = d2; d2 = d1; d1 = cur; cur = tmp;
    }
    wg_barrier_wait();                               // consume final round
}

__global__ __launch_bounds__(256) void dtw_pi_kernel(const float* __restrict__ mu,
                                                     const float* __restrict__ mask,
                                                     float* __restrict__ pi) {
    int idx = blockIdx.x * 256 + threadIdx.x;
    const int total = BATCH * NA * NB;
    if (idx >= total) return;
    int j    = idx % NB;
    int rest = idx / NB;
    int i    = rest % NA;
    int b    = rest / NA;
    const float* mub = mu + (size_t)b * MU_H * MU_W;

    float up   = mub[(size_t)i * MU_W + (j + 1)];    // mu[:-1, 1:]
    float left = mub[(size_t)(i + 1) * MU_W + j];    // mu[1:, :-1]
    float dg   = mub[(size_t)i * MU_W + j];          // mu[:-1, :-1]

    // softmax over moves; the alpha*W broadcast term cancels exactly.
    float m  = fmaxf(up, fmaxf(left, dg));
    float e0 = __expf(up - m);
    float e1 = __expf(left - m);
    float e2 = __expf(dg - m);
    float msk = __builtin_nontemporal_load(mask + idx);   // read-once stream
    float scale = msk / (e0 + e1 + e2);

    size_t o = (size_t)idx * 3;                      // write-once stream: keep out of L2
    __builtin_nontemporal_store(e0 * scale, pi + o + 0);
    __builtin_nontemporal_store(e1 * scale, pi + o + 1);
    __builtin_nontemporal_store(e2 * scale, pi + o + 2);
}

extern "C" void kernel_launch(void* const* d_in, const int* in_sizes, int n_in,
                              void* d_out, int out_size, void* d_ws, size_t ws_size,
                              hipStream_t stream) {
    const float* W    = (const float*)d_in[0];   // [64,384,384] f32
    const float* mask = (const float*)d_in[1];   // [64,384,384,1] f32
    float* mu = (float*)d_out;                                   // [64,385,385]
    float* pi = mu + (size_t)BATCH * MU_H * MU_W;                // [64,384,384,3]

    dtw_mu_kernel<<<dim3(BATCH), dim3(NA), 0, stream>>>(W, mu);

    int total = BATCH * NA * NB;
    dtw_pi_kernel<<<dim3((total + 255) / 256), dim3(256), 0, stream>>>(mu, mask, pi);
}